// LSTMClassifier_16707422781615
// MI455X (gfx1250) — compile-verified
//
#include <hip/hip_runtime.h>
#include <math.h>

// LSTM classifier: B=256, S=384, D=192, H=100, 4H=400, C=250
// Stage 1: pre[s][b][0:400] = x[b][s][:] @ W + bias      (parallel GEMM, WMMA f32)
// Stage 2: sequential LSTM scan, 16 batch rows / workgroup (WMMA f32 for h@U)
//          - U cached in LDS (156.25KB of the 320KB/WGP budget; only 16 WGs exist)
//          - pre[s+1] prefetched (global_prefetch_b8) while step s computes
// Stage 3: logits = hT @ Wd + bd                          (WMMA f32)
// Workspace: pre = 384*256*400 f32 (157.3MB), hT = 256*100 f32.

typedef __attribute__((ext_vector_type(2))) float v2f;
typedef __attribute__((ext_vector_type(8))) float v8f;

#define S_LEN 384
#define B_SZ  256
#define D_IN  192
#define H_DIM 100
#define FH    400
#define C_OUT 250

#define XS_STRIDE 200   // LDS stride for staged x tile (>=192)
#define HS  105         // LDS stride for h (odd -> conflict-free column reads)
#define ZS  404         // LDS stride for z

__device__ __forceinline__ float sigmoidf_(float v) {
    return 1.0f / (1.0f + __expf(-v));
}

__device__ __forceinline__ float fast_tanhf_(float v) {
#if __has_builtin(__builtin_amdgcn_tanhf)
    return __builtin_amdgcn_tanhf(v);        // v_tanh_f32 (gfx1250 TRANS op)
#elif __has_builtin(__builtin_amdgcn_tanh_f32)
    return __builtin_amdgcn_tanh_f32(v);
#else
    return tanhf(v);
#endif
}

// ---------------- Stage 1: input projection ----------------
// grid = S * (B/16) blocks, 256 threads (8 waves). Each block: one 16-row
// (fixed s, 16 consecutive b) M-tile; waves split the 25 N-tiles of 16.
__global__ __launch_bounds__(256)
void lstm_xw_kernel(const float* __restrict__ x, const float* __restrict__ W,
                    const float* __restrict__ bias, float* __restrict__ pre) {
    __shared__ float xs[16 * XS_STRIDE];
    const int blk = blockIdx.x;
    const int s   = blk >> 4;          // / (B/16)
    const int b0  = (blk & 15) << 4;
    const int tid = threadIdx.x;

    // Stage the 16x192 x-tile into LDS (rows of x are S*D apart per batch).
    for (int e = tid; e < 16 * D_IN; e += 256) {
        const int r = e / D_IN, d = e - r * D_IN;
        xs[r * XS_STRIDE + d] = x[((size_t)(b0 + r) * S_LEN + s) * D_IN + d];
    }
    __syncthreads();

    const int wave  = tid >> 5;
    const int lane  = tid & 31;
    const int lhalf = lane >> 4;       // 0: K pair (k0,k0+1), 1: (k0+2,k0+3)
    const int l15   = lane & 15;
    const int koff  = lhalf * 2;

    for (int t = wave; t < 25; t += 8) {
        const int n0 = t << 4;
        const int n  = n0 + l15;
        const float bb = bias[n];
        v8f acc;
#pragma unroll
        for (int r = 0; r < 8; ++r) acc[r] = bb;   // bias broadcast across M

#pragma unroll 4
        for (int kk = 0; kk < 48; ++kk) {          // K = 192 = 48 * 4
            const int k0 = kk * 4 + koff;
            v2f a, bf;
            a[0]  = xs[l15 * XS_STRIDE + k0];
            a[1]  = xs[l15 * XS_STRIDE + k0 + 1];
            bf[0] = W[(size_t)k0 * FH + n];
            bf[1] = W[(size_t)(k0 + 1) * FH + n];
            acc = __builtin_amdgcn_wmma_f32_16x16x4_f32(
                false, a, false, bf, (short)0, acc, false, false);
        }
        float* prow = pre + ((size_t)s * B_SZ + b0) * FH;
#pragma unroll
        for (int r = 0; r < 8; ++r)
            prow[(size_t)(r + lhalf * 8) * FH + n] = acc[r];
    }
}

// ---------------- Stage 2: sequential recurrence ----------------
// grid = B/16 = 16 blocks, 256 threads. Each block owns 16 batch rows for the
// entire time loop; U, h, c, z live in LDS. Per step: z = pre[s] + h @ U
// (WMMA f32), then elementwise gates. Latency-bound -> U in LDS kills the L2
// round-trip inside the 25-deep WMMA K-chain; prefetch hides the pre loads.
__global__ __launch_bounds__(256)
void lstm_rec_kernel(const float* __restrict__ pre, const float* __restrict__ U,
                     float* __restrict__ hT) {
    __shared__ float U_lds[H_DIM * FH];   // 156.25 KB (<320KB/WGP; 16 WGs total)
    __shared__ float h_s[16 * HS];
    __shared__ float c_s[16 * H_DIM];
    __shared__ float z_s[16 * ZS];

    const int b0  = blockIdx.x << 4;
    const int tid = threadIdx.x;
    const int wave  = tid >> 5;
    const int lane  = tid & 31;
    const int lhalf = lane >> 4;
    const int l15   = lane & 15;
    const int koff  = lhalf * 2;

    for (int e = tid; e < H_DIM * FH; e += 256) U_lds[e] = U[e];
    for (int e = tid; e < 16 * HS; e += 256)    h_s[e] = 0.0f;
    for (int e = tid; e < 16 * H_DIM; e += 256) c_s[e] = 0.0f;
    __syncthreads();

    for (int s = 0; s < S_LEN; ++s) {
        const float* prow = pre + ((size_t)s * B_SZ + b0) * FH;

        // Prefetch next step's pre tile (16 rows x 1600B = 400 cachelines).
        if (s + 1 < S_LEN) {
            const float* nrow = prow + (size_t)B_SZ * FH;
            for (int e = tid; e < 400; e += 256)
                __builtin_prefetch(nrow + e * 16, 0, 0);
        }

        // z = pre + h @ U : waves split the 25 N-tiles
        for (int t = wave; t < 25; t += 8) {
            const int n0 = t << 4;
            const int n  = n0 + l15;
            v8f acc;
#pragma unroll
            for (int r = 0; r < 8; ++r)        // seed accumulator with pre
                acc[r] = prow[(size_t)(r + lhalf * 8) * FH + n];

#pragma unroll
            for (int kk = 0; kk < 25; ++kk) {  // K = 100 = 25 * 4
                const int k0 = kk * 4 + koff;
                v2f a, bf;
                a[0]  = h_s[l15 * HS + k0];
                a[1]  = h_s[l15 * HS + k0 + 1];
                bf[0] = U_lds[k0 * FH + n];
                bf[1] = U_lds[(k0 + 1) * FH + n];
                acc = __builtin_amdgcn_wmma_f32_16x16x4_f32(
                    false, a, false, bf, (short)0, acc, false, false);
            }
#pragma unroll
            for (int r = 0; r < 8; ++r)
                z_s[(r + lhalf * 8) * ZS + n] = acc[r];
        }
        __syncthreads();

        // Gates: i,f,g,o at offsets 0,100,200,300 (Keras order)
        for (int e = tid; e < 16 * H_DIM; e += 256) {
            const int b = e / H_DIM, j = e - b * H_DIM;
            const float zi = z_s[b * ZS + j];
            const float zf = z_s[b * ZS + H_DIM + j];
            const float zg = z_s[b * ZS + 2 * H_DIM + j];
            const float zo = z_s[b * ZS + 3 * H_DIM + j];
            const float ig = sigmoidf_(zi);
            const float fg = sigmoidf_(zf);
            const float gg = fast_tanhf_(zg);
            const float og = sigmoidf_(zo);
            const float cn = fg * c_s[e] + ig * gg;
            c_s[e] = cn;
            h_s[b * HS + j] = og * fast_tanhf_(cn);
        }
        __syncthreads();
    }

    for (int e = tid; e < 16 * H_DIM; e += 256) {
        const int b = e / H_DIM, j = e - b * H_DIM;
        hT[(size_t)(b0 + b) * H_DIM + j] = h_s[b * HS + j];
    }
}

// ---------------- Stage 3: dense head ----------------
// grid = (B/16) * ceil(C/16) = 16*16 blocks of 1 wave; ragged N=250 guarded.
__global__ __launch_bounds__(32)
void dense_kernel(const float* __restrict__ hT, const float* __restrict__ Wd,
                  const float* __restrict__ bd, float* __restrict__ out) {
    const int m0 = (blockIdx.x >> 4) << 4;
    const int n0 = (blockIdx.x & 15) << 4;
    const int lane  = threadIdx.x & 31;
    const int lhalf = lane >> 4;
    const int l15   = lane & 15;
    const int koff  = lhalf * 2;
    const int n = n0 + l15;
    const bool nvalid = (n < C_OUT);

    v8f acc;
    const float bb = nvalid ? bd[n] : 0.0f;
#pragma unroll
    for (int r = 0; r < 8; ++r) acc[r] = bb;

#pragma unroll
    for (int kk = 0; kk < 25; ++kk) {          // K = 100
        const int k0 = kk * 4 + koff;
        v2f a, bf;
        a[0]  = hT[(size_t)(m0 + l15) * H_DIM + k0];
        a[1]  = hT[(size_t)(m0 + l15) * H_DIM + k0 + 1];
        bf[0] = nvalid ? Wd[(size_t)k0 * C_OUT + n] : 0.0f;
        bf[1] = nvalid ? Wd[(size_t)(k0 + 1) * C_OUT + n] : 0.0f;
        acc = __builtin_amdgcn_wmma_f32_16x16x4_f32(
            false, a, false, bf, (short)0, acc, false, false);
    }
    if (nvalid) {
#pragma unroll
        for (int r = 0; r < 8; ++r)
            out[(size_t)(m0 + r + lhalf * 8) * C_OUT + n] = acc[r];
    }
}

extern "C" void kernel_launch(void* const* d_in, const int* in_sizes, int n_in,
                              void* d_out, int out_size, void* d_ws, size_t ws_size,
                              hipStream_t stream) {
    const float* x  = (const float*)d_in[0];  // [256,384,12,16] -> [256][384][192]
    const float* W  = (const float*)d_in[1];  // [192][400]
    const float* U  = (const float*)d_in[2];  // [100][400]
    const float* bi = (const float*)d_in[3];  // [400]
    const float* Wd = (const float*)d_in[4];  // [100][250]
    const float* bd = (const float*)d_in[5];  // [250]
    float* out = (float*)d_out;               // [256][250]

    float* pre = (float*)d_ws;                            // 384*256*400 f32
    float* hT  = pre + (size_t)S_LEN * B_SZ * FH;         // 256*100 f32

    lstm_xw_kernel<<<S_LEN * (B_SZ / 16), 256, 0, stream>>>(x, W, bi, pre);
    lstm_rec_kernel<<<B_SZ / 16, 256, 0, stream>>>(pre, U, hT);
    dense_kernel<<<(B_SZ / 16) * 16, 32, 0, stream>>>(hT, Wd, bd, out);
}